// BaseGAT_45337674776791
// MI455X (gfx1250) — compile-verified
//
#include <hip/hip_runtime.h>
#include <hip/hip_bf16.h>
#include <math.h>

// ---------------- problem constants (fixed by the reference) ----------------
#define NPT   32000
#define NNODE (3 * NPT)          // 96000
#define NEDGE (NNODE * 16)       // 1,536,000
#define NHEAD 4
#define DDIM  8
#define HD    (NHEAD * DDIM)     // 32
#define NEG_SLOPE 0.2f

typedef __attribute__((ext_vector_type(2))) float v2f;
typedef __attribute__((ext_vector_type(8))) float v8f;

// ---------------------------------------------------------------------------
// Per-type embed GEMM:  T[row_offset + m, :32] = A[m,:K] @ W[:K,:32] + bias
// One wave32 computes a 16x32 output slab via two V_WMMA_F32_16X16X4_F32
// accumulators, looping K in steps of 4.
// A fragment (16x4 f32): lanes 0-15 hold M=0..15, VGPR0=K0/VGPR1=K1;
//                        lanes 16-31 hold K2/K3.   (ISA 7.12.2)
// B fragment (4x16 f32): mirrored — lane%16 = N, lane/16 selects K pair.
// C/D (16x16 f32): 8 VGPRs; lane%16 = N, VGPR v = M=v (lanes<16) / M=v+8.
// ---------------------------------------------------------------------------
__global__ void __launch_bounds__(256)
gemm_embed_wmma(const float* __restrict__ A, const float* __restrict__ W,
                const float* __restrict__ bias, float* __restrict__ T,
                int M, int K, int row_offset) {
  const int wave = (blockIdx.x * blockDim.x + threadIdx.x) >> 5;
  const int lane = threadIdx.x & 31;
  const int m0   = wave * 16;
  if (m0 >= M) return;                 // wave-uniform: EXEC stays all-ones

  const int half = lane >> 4;          // 0: K pair {0,1} / M 0-7 ; 1: {2,3} / M 8-15
  const int r    = lane & 15;          // M row (A) or N col (B, C/D)

  v8f c0 = {};   // output cols 0..15
  v8f c1 = {};   // output cols 16..31

  const float* arow = A + (size_t)(m0 + r) * K;
  for (int k = 0; k < K; k += 4) {
    const int ka = k + 2 * half;
    v2f a;  a.x  = arow[ka];                 a.y  = arow[ka + 1];
    v2f b0; b0.x = W[(size_t)ka * HD + r];       b0.y = W[(size_t)(ka + 1) * HD + r];
    v2f b1; b1.x = W[(size_t)ka * HD + 16 + r];  b1.y = W[(size_t)(ka + 1) * HD + 16 + r];
    c0 = __builtin_amdgcn_wmma_f32_16x16x4_f32(false, a, false, b0, (short)0, c0, false, false);
    c1 = __builtin_amdgcn_wmma_f32_16x16x4_f32(false, a, false, b1, (short)0, c1, false, false);
  }

  const float bl = bias[r];
  const float bh = bias[16 + r];
#pragma unroll
  for (int v = 0; v < 8; ++v) {
    const int row = row_offset + m0 + v + 8 * half;
    T[(size_t)row * HD + r]      = c0[v] + bl;
    T[(size_t)row * HD + 16 + r] = c1[v] + bh;
  }
}

// ---------------- init: zero out accumulator, m=-inf, s=0 ------------------
__global__ void init_out(float* __restrict__ out, int n) {
  int i = blockIdx.x * blockDim.x + threadIdx.x;
  if (i < n) out[i] = 0.0f;
}
__global__ void init_ms(float* __restrict__ m, float* __restrict__ s, int n) {
  int i = blockIdx.x * blockDim.x + threadIdx.x;
  if (i < n) { m[i] = -INFINITY; s[i] = 0.0f; }
}

// ---------------- el / er projections: [N,H] dots over D=8 -----------------
__global__ void attn_proj(const float* __restrict__ T,
                          const float* __restrict__ al,
                          const float* __restrict__ ar,
                          float* __restrict__ el, float* __restrict__ er, int n) {
  int i = blockIdx.x * blockDim.x + threadIdx.x;   // i = node*H + head
  if (i >= n) return;
  const int head = i & (NHEAD - 1);
  const float* hv = T + (size_t)(i >> 2) * HD + head * DDIM;
  float sl = 0.0f, sr = 0.0f;
#pragma unroll
  for (int d = 0; d < DDIM; ++d) {
    const float h = hv[d];
    sl += h * al[head * DDIM + d];
    sr += h * ar[head * DDIM + d];
  }
  el[i] = sl; er[i] = sr;
}

// ---------------- float atomic max (sign-split trick) ----------------------
__device__ __forceinline__ void atomicMaxF32(float* addr, float v) {
  if (v >= 0.0f) atomicMax((int*)addr, __float_as_int(v));
  else           atomicMin((unsigned int*)addr, __float_as_uint(v));
}

__device__ __forceinline__ float leaky(float x) {
  return x > 0.0f ? x : NEG_SLOPE * x;
}

// ---------------- edge pass A: segment max of leaky logits -----------------
__global__ void edge_max(const float* __restrict__ el, const float* __restrict__ er,
                         const int* __restrict__ src, const int* __restrict__ dst,
                         float* __restrict__ m, int eh) {
  int i = blockIdx.x * blockDim.x + threadIdx.x;   // i = edge*H + head
  if (i >= eh) return;
  const int e = i >> 2, head = i & (NHEAD - 1);
  const int sN = src[e], dN = dst[e];
  const float v = leaky(el[sN * NHEAD + head] + er[dN * NHEAD + head]);
  atomicMaxF32(&m[dN * NHEAD + head], v);
}

// ---------------- edge pass B: segment sum of exp(e - m[dst]) --------------
__global__ void edge_expsum(const float* __restrict__ el, const float* __restrict__ er,
                            const int* __restrict__ src, const int* __restrict__ dst,
                            const float* __restrict__ m, float* __restrict__ s, int eh) {
  int i = blockIdx.x * blockDim.x + threadIdx.x;
  if (i >= eh) return;
  const int e = i >> 2, head = i & (NHEAD - 1);
  const int sN = src[e], dN = dst[e];
  const float v  = leaky(el[sN * NHEAD + head] + er[dN * NHEAD + head]);
  const float ex = __expf(v - m[dN * NHEAD + head]);
  atomicAdd(&s[dN * NHEAD + head], ex);
}

// ---------------- edge pass C: alpha-weighted message scatter --------------
__global__ void edge_aggregate(const float* __restrict__ el, const float* __restrict__ er,
                               const int* __restrict__ src, const int* __restrict__ dst,
                               const float* __restrict__ m, const float* __restrict__ s,
                               const float* __restrict__ T, float* __restrict__ out, int eh) {
  int i = blockIdx.x * blockDim.x + threadIdx.x;
  if (i >= eh) return;
  const int e = i >> 2, head = i & (NHEAD - 1);
  const int sN = src[e], dN = dst[e];
  const float v     = leaky(el[sN * NHEAD + head] + er[dN * NHEAD + head]);
  const float ex    = __expf(v - m[dN * NHEAD + head]);
  const float alpha = ex / (s[dN * NHEAD + head] + 1e-9f);
  const float* hv = T   + (size_t)sN * HD + head * DDIM;
  float*       ov = out + (size_t)dN * HD + head * DDIM;
#pragma unroll
  for (int d = 0; d < DDIM; ++d) atomicAdd(&ov[d], alpha * hv[d]);
}

// ---------------- final ELU in place ---------------------------------------
__global__ void elu_inplace(float* __restrict__ out, int n) {
  int i = blockIdx.x * blockDim.x + threadIdx.x;
  if (i >= n) return;
  const float x = out[i];
  out[i] = x > 0.0f ? x : expm1f(x);
}

// ---------------------------------------------------------------------------
extern "C" void kernel_launch(void* const* d_in, const int* in_sizes, int n_in,
                              void* d_out, int out_size, void* d_ws, size_t ws_size,
                              hipStream_t stream) {
  // setup_inputs() dict order:
  const float* x0     = (const float*)d_in[0];
  const float* x1     = (const float*)d_in[1];
  const float* x2     = (const float*)d_in[2];
  const float* W0     = (const float*)d_in[3];
  const float* b0     = (const float*)d_in[4];
  const float* W1     = (const float*)d_in[5];
  const float* b1     = (const float*)d_in[6];
  const float* W2     = (const float*)d_in[7];
  const float* b2     = (const float*)d_in[8];
  const float* attn_l = (const float*)d_in[9];
  const float* attn_r = (const float*)d_in[10];
  // d_in[11] = type_mask, d_in[12..14] = idx0..idx2 (contiguous blocks, implicit)
  const int*   src    = (const int*)d_in[15];
  const int*   dst    = (const int*)d_in[16];
  float* out = (float*)d_out;

  // workspace: T (N*32) | el (N*4) | er (N*4) | m (N*4) | s (N*4)  -> ~18.4 MB
  float* T  = (float*)d_ws;
  float* el = T  + (size_t)NNODE * HD;
  float* er = el + (size_t)NNODE * NHEAD;
  float* m  = er + (size_t)NNODE * NHEAD;
  float* s  = m  + (size_t)NNODE * NHEAD;

  const int TPB = 256;

  // init accumulators
  init_out<<<(NNODE * HD + TPB - 1) / TPB, TPB, 0, stream>>>(out, NNODE * HD);
  init_ms<<<(NNODE * NHEAD + TPB - 1) / TPB, TPB, 0, stream>>>(m, s, NNODE * NHEAD);

  // per-type WMMA embeds into T (types occupy contiguous row blocks)
  const int waves = NPT / 16;               // 2000 tiles per type
  const int blocks = (waves * 32 + TPB - 1) / TPB;
  gemm_embed_wmma<<<blocks, TPB, 0, stream>>>(x0, W0, b0, T, NPT, 128, 0 * NPT);
  gemm_embed_wmma<<<blocks, TPB, 0, stream>>>(x1, W1, b1, T, NPT,  64, 1 * NPT);
  gemm_embed_wmma<<<blocks, TPB, 0, stream>>>(x2, W2, b2, T, NPT,  32, 2 * NPT);

  // attention projections
  attn_proj<<<(NNODE * NHEAD + TPB - 1) / TPB, TPB, 0, stream>>>(T, attn_l, attn_r,
                                                                 el, er, NNODE * NHEAD);

  // edge passes
  const int EH = NEDGE * NHEAD;
  edge_max<<<(EH + TPB - 1) / TPB, TPB, 0, stream>>>(el, er, src, dst, m, EH);
  edge_expsum<<<(EH + TPB - 1) / TPB, TPB, 0, stream>>>(el, er, src, dst, m, s, EH);
  edge_aggregate<<<(EH + TPB - 1) / TPB, TPB, 0, stream>>>(el, er, src, dst, m, s, T, out, EH);

  // final activation
  elu_inplace<<<(NNODE * HD + TPB - 1) / TPB, TPB, 0, stream>>>(out, NNODE * HD);
}